// GAT_8323646620578
// MI455X (gfx1250) — compile-verified
//
#include <hip/hip_runtime.h>
#include <math.h>

// ---------------------------------------------------------------------------
// GAT 2-layer forward for MI455X (gfx1250, wave32).
// GEMMs use V_WMMA_F32_16X16X4_F32 (exact fp32 matrix path); each wave owns a
// 16-row output stripe with all column tiles as accumulators so the A stream
// (the dominant GEMM traffic) is loaded exactly once.
// Edge softmax/scatter passes are flat atomic kernels; all node-state buffers
// (h1 12.8MB, out1 12.8MB, h2 8MB) are L2-resident on the 192MB L2.
// ---------------------------------------------------------------------------

typedef __attribute__((ext_vector_type(2))) float v2f;
typedef __attribute__((ext_vector_type(8))) float v8f;

#define GAT_NEG_SLOPE 0.2f
#define GAT_EPS 1e-16f

#define WMMA_F32(A, B, C) \
  __builtin_amdgcn_wmma_f32_16x16x4_f32(false, (A), false, (B), (short)0, (C), false, false)

__device__ __forceinline__ void atomic_max_float(float* addr, float val) {
  // Totally ordered float max via int/uint atomics (valid with -inf init).
  if (val >= 0.0f)
    atomicMax((int*)addr, __float_as_int(val));
  else
    atomicMin((unsigned int*)addr, (unsigned int)__float_as_int(val));
}

__global__ void gat_fill(float* __restrict__ p, float v, int n) {
  int i = blockIdx.x * blockDim.x + threadIdx.x;
  if (i < n) p[i] = v;
}

// ---------------- GEMM1: h1[N,64] = x[N,128] @ W1[128,64]  (WMMA f32) -------
// One wave per 16-row stripe; 4 column tiles held as accumulators.
__global__ void gat_gemm1_wmma(const float* __restrict__ x,
                               const float* __restrict__ W1,
                               float* __restrict__ h1, int N) {
  const int lane = threadIdx.x;                       // 0..31
  const int row0 = (blockIdx.x * 4 + threadIdx.y) * 16;
  const int m    = lane & 15;
  const int half = lane >> 4;                         // 0: K=k,k+1 ; 1: K=k+2,k+3
  const int rowA = min(row0 + m, N - 1);              // clamp keeps EXEC full
  const float* arow = x + (size_t)rowA * 128;

  v8f c0 = {}, c1 = {}, c2 = {}, c3 = {};
  for (int k = 0; k < 128; k += 4) {
    const int ka = k + 2 * half;
    v2f a; a.x = arow[ka]; a.y = arow[ka + 1];
    const float* b0p = W1 + (size_t)ka * 64 + m;
    const float* b1p = W1 + (size_t)(ka + 1) * 64 + m;
    v2f b;
    b.x = b0p[0];  b.y = b1p[0];  c0 = WMMA_F32(a, b, c0);
    b.x = b0p[16]; b.y = b1p[16]; c1 = WMMA_F32(a, b, c1);
    b.x = b0p[32]; b.y = b1p[32]; c2 = WMMA_F32(a, b, c2);
    b.x = b0p[48]; b.y = b1p[48]; c3 = WMMA_F32(a, b, c3);
  }

  if (row0 + 16 <= N) {                               // uniform fast path
#pragma unroll
    for (int r = 0; r < 8; ++r) {
      float* p = h1 + (size_t)(row0 + half * 8 + r) * 64 + m;
      p[0] = c0[r]; p[16] = c1[r]; p[32] = c2[r]; p[48] = c3[r];
    }
  } else {
#pragma unroll
    for (int r = 0; r < 8; ++r) {
      const int row = row0 + half * 8 + r;
      if (row < N) {
        float* p = h1 + (size_t)row * 64 + m;
        p[0] = c0[r]; p[16] = c1[r]; p[32] = c2[r]; p[48] = c3[r];
      }
    }
  }
}

// ---------------- GEMM2: h2[N,40] = hact[N,64] @ W2[64,40]  (WMMA f32) ------
// One wave per 16-row stripe; 3 column tiles (cols padded 40 -> 48).
__global__ void gat_gemm2_wmma(const float* __restrict__ hact,
                               const float* __restrict__ W2,
                               float* __restrict__ h2, int N) {
  const int lane = threadIdx.x;
  const int row0 = (blockIdx.x * 4 + threadIdx.y) * 16;
  const int m    = lane & 15;
  const int half = lane >> 4;
  const int rowA = min(row0 + m, N - 1);
  const float* arow = hact + (size_t)rowA * 64;
  const int  col2   = min(32 + m, 39);                // clamped address
  const bool col2ok = (32 + m) < 40;                  // select, EXEC stays full

  v8f c0 = {}, c1 = {}, c2 = {};
  for (int k = 0; k < 64; k += 4) {
    const int ka = k + 2 * half;
    v2f a; a.x = arow[ka]; a.y = arow[ka + 1];
    const float* b0p = W2 + (size_t)ka * 40;
    const float* b1p = W2 + (size_t)(ka + 1) * 40;
    v2f b;
    b.x = b0p[m];      b.y = b1p[m];      c0 = WMMA_F32(a, b, c0);
    b.x = b0p[16 + m]; b.y = b1p[16 + m]; c1 = WMMA_F32(a, b, c1);
    const float bx2 = b0p[col2], by2 = b1p[col2];
    b.x = col2ok ? bx2 : 0.0f; b.y = col2ok ? by2 : 0.0f;
    c2 = WMMA_F32(a, b, c2);
  }

  if (row0 + 16 <= N) {                               // uniform fast path
#pragma unroll
    for (int r = 0; r < 8; ++r) {
      float* p = h2 + (size_t)(row0 + half * 8 + r) * 40;
      p[m] = c0[r]; p[16 + m] = c1[r];
      if (col2ok) p[32 + m] = c2[r];
    }
  } else {
#pragma unroll
    for (int r = 0; r < 8; ++r) {
      const int row = row0 + half * 8 + r;
      if (row < N) {
        float* p = h2 + (size_t)row * 40;
        p[m] = c0[r]; p[16 + m] = c1[r];
        if (col2ok) p[32 + m] = c2[r];
      }
    }
  }
}

// ---------------- Layer-1 attention coefficients ----------------------------
__global__ void gat_att1(const float* __restrict__ h1,
                         const float* __restrict__ att_src,
                         const float* __restrict__ att_dst,
                         float* __restrict__ a_src, float* __restrict__ a_dst,
                         int N) {
  int t = blockIdx.x * blockDim.x + threadIdx.x;
  if (t >= N * 8) return;
  const int n = t >> 3, hh = t & 7;
  const float* hp = h1 + (size_t)n * 64 + hh * 8;
  const float* as = att_src + hh * 8;
  const float* ad = att_dst + hh * 8;
  float s = 0.0f, d = 0.0f;
#pragma unroll
  for (int c = 0; c < 8; ++c) { float v = hp[c]; s += v * as[c]; d += v * ad[c]; }
  a_src[t] = s;
  a_dst[t] = d;
}

// ---------------- Layer-1 edge passes ---------------------------------------
__global__ void gat_edge_max1(const long long* __restrict__ ei,
                              const float* __restrict__ a_src,
                              const float* __restrict__ a_dst,
                              float* __restrict__ emax, int E) {
  int t = blockIdx.x * blockDim.x + threadIdx.x;
  if (t >= E * 8) return;
  const int e = t >> 3, hh = t & 7;
  const long long s = ei[e], d = ei[(size_t)E + e];
  float v = a_src[s * 8 + hh] + a_dst[d * 8 + hh];
  v = (v >= 0.0f) ? v : GAT_NEG_SLOPE * v;
  atomic_max_float(&emax[d * 8 + hh], v);
}

__global__ void gat_edge_exp1(const long long* __restrict__ ei,
                              const float* __restrict__ a_src,
                              const float* __restrict__ a_dst,
                              const float* __restrict__ emax,
                              float* __restrict__ ex,
                              float* __restrict__ denom, int E) {
  int t = blockIdx.x * blockDim.x + threadIdx.x;
  if (t >= E * 8) return;
  const int e = t >> 3, hh = t & 7;
  const long long s = ei[e], d = ei[(size_t)E + e];
  float v = a_src[s * 8 + hh] + a_dst[d * 8 + hh];
  v = (v >= 0.0f) ? v : GAT_NEG_SLOPE * v;
  const float xv = expf(v - emax[d * 8 + hh]);
  ex[t] = xv;
  atomicAdd(&denom[d * 8 + hh], xv);
}

__global__ void gat_edge_scatter1(const long long* __restrict__ ei,
                                  const float* __restrict__ ex,
                                  const float* __restrict__ denom,
                                  const float* __restrict__ h1,
                                  float* __restrict__ out1, int E) {
  int t = blockIdx.x * blockDim.x + threadIdx.x;
  if (t >= E * 8) return;
  const int e = t >> 3, hh = t & 7;
  const long long s = ei[e], d = ei[(size_t)E + e];
  const float alpha = ex[t] / (denom[d * 8 + hh] + GAT_EPS);
  const float* hp = h1 + (size_t)s * 64 + hh * 8;
  float* op = out1 + (size_t)d * 64 + hh * 8;
#pragma unroll
  for (int c = 0; c < 8; ++c) atomicAdd(&op[c], alpha * hp[c]);
}

// ---------------- ELU(out1 + bias1), in place -------------------------------
__global__ void gat_elu_bias(float* __restrict__ out1,
                             const float* __restrict__ bias1, int N) {
  int t = blockIdx.x * blockDim.x + threadIdx.x;
  if (t >= N * 64) return;
  const float v = out1[t] + bias1[t & 63];
  out1[t] = (v > 0.0f) ? v : (expf(v) - 1.0f);
}

// ---------------- Layer-2 attention coefficients ----------------------------
__global__ void gat_att2(const float* __restrict__ h2,
                         const float* __restrict__ att_src,
                         const float* __restrict__ att_dst,
                         float* __restrict__ a_src, float* __restrict__ a_dst,
                         int N) {
  int n = blockIdx.x * blockDim.x + threadIdx.x;
  if (n >= N) return;
  const float* hp = h2 + (size_t)n * 40;
  float s = 0.0f, d = 0.0f;
#pragma unroll
  for (int c = 0; c < 40; ++c) { float v = hp[c]; s += v * att_src[c]; d += v * att_dst[c]; }
  a_src[n] = s;
  a_dst[n] = d;
}

// ---------------- Layer-2 edge passes ---------------------------------------
__global__ void gat_edge_max2(const long long* __restrict__ ei,
                              const float* __restrict__ a_src,
                              const float* __restrict__ a_dst,
                              float* __restrict__ emax, int E) {
  int e = blockIdx.x * blockDim.x + threadIdx.x;
  if (e >= E) return;
  const long long s = ei[e], d = ei[(size_t)E + e];
  float v = a_src[s] + a_dst[d];
  v = (v >= 0.0f) ? v : GAT_NEG_SLOPE * v;
  atomic_max_float(&emax[d], v);
}

__global__ void gat_edge_exp2(const long long* __restrict__ ei,
                              const float* __restrict__ a_src,
                              const float* __restrict__ a_dst,
                              const float* __restrict__ emax,
                              float* __restrict__ ex,
                              float* __restrict__ denom, int E) {
  int e = blockIdx.x * blockDim.x + threadIdx.x;
  if (e >= E) return;
  const long long s = ei[e], d = ei[(size_t)E + e];
  float v = a_src[s] + a_dst[d];
  v = (v >= 0.0f) ? v : GAT_NEG_SLOPE * v;
  const float xv = expf(v - emax[d]);
  ex[e] = xv;
  atomicAdd(&denom[d], xv);
}

__global__ void gat_edge_scatter2(const long long* __restrict__ ei,
                                  const float* __restrict__ ex,
                                  const float* __restrict__ denom,
                                  const float* __restrict__ h2,
                                  float* __restrict__ out, int E) {
  int t = blockIdx.x * blockDim.x + threadIdx.x;
  if (t >= E * 5) return;               // 5 chunks of 8 channels = 40
  const int e = t / 5, ch = t - e * 5;
  const long long s = ei[e], d = ei[(size_t)E + e];
  const float alpha = ex[e] / (denom[d] + GAT_EPS);
  const int c0 = ch * 8;
  const float* hp = h2 + (size_t)s * 40 + c0;
  float* op = out + (size_t)d * 40 + c0;
#pragma unroll
  for (int c = 0; c < 8; ++c) atomicAdd(&op[c], alpha * hp[c]);
}

__global__ void gat_bias2(float* __restrict__ out,
                          const float* __restrict__ bias2, int N) {
  int t = blockIdx.x * blockDim.x + threadIdx.x;
  if (t >= N * 40) return;
  out[t] += bias2[t % 40];
}

// ---------------------------------------------------------------------------
extern "C" void kernel_launch(void* const* d_in, const int* in_sizes, int n_in,
                              void* d_out, int out_size, void* d_ws, size_t ws_size,
                              hipStream_t stream) {
  const float*     x        = (const float*)d_in[0];
  const long long* ei       = (const long long*)d_in[1];   // int64 [2,E]
  const float*     W1       = (const float*)d_in[2];
  const float*     att_src1 = (const float*)d_in[3];
  const float*     att_dst1 = (const float*)d_in[4];
  const float*     bias1    = (const float*)d_in[5];
  const float*     W2       = (const float*)d_in[6];
  const float*     att_src2 = (const float*)d_in[7];
  const float*     att_dst2 = (const float*)d_in[8];
  const float*     bias2    = (const float*)d_in[9];
  float*           out      = (float*)d_out;

  const int N = in_sizes[0] / 128;
  const int E = in_sizes[1] / 2;

  // Workspace partition (~70 MB total)
  float* wp = (float*)d_ws;
  auto take = [&](size_t n) { float* p = wp; wp += n; return p; };
  float* h1    = take((size_t)N * 64);
  float* asrc1 = take((size_t)N * 8);
  float* adst1 = take((size_t)N * 8);
  float* emax1 = take((size_t)N * 8);
  float* den1  = take((size_t)N * 8);
  float* out1  = take((size_t)N * 64);   // layer-1 aggregate, then hact (ELU)
  float* ex1   = take((size_t)E * 8);
  float* h2    = take((size_t)N * 40);
  float* asrc2 = take((size_t)N);
  float* adst2 = take((size_t)N);
  float* emax2 = take((size_t)N);
  float* den2  = take((size_t)N);
  float* ex2   = take((size_t)E);

  const int B = 256;
  auto nb = [](int n, int b) { return (n + b - 1) / b; };
  const float NINF = -__builtin_inff();

  // Init accumulators (re-done every call: deterministic under graph replay)
  gat_fill<<<nb(N * 8, B), B, 0, stream>>>(emax1, NINF, N * 8);
  gat_fill<<<nb(N * 8, B), B, 0, stream>>>(den1, 0.0f, N * 8);
  gat_fill<<<nb(N * 64, B), B, 0, stream>>>(out1, 0.0f, N * 64);
  gat_fill<<<nb(N, B), B, 0, stream>>>(emax2, NINF, N);
  gat_fill<<<nb(N, B), B, 0, stream>>>(den2, 0.0f, N);
  gat_fill<<<nb(N * 40, B), B, 0, stream>>>(out, 0.0f, N * 40);

  // Layer 1
  gat_gemm1_wmma<<<dim3((N + 63) / 64), dim3(32, 4), 0, stream>>>(x, W1, h1, N);
  gat_att1<<<nb(N * 8, B), B, 0, stream>>>(h1, att_src1, att_dst1, asrc1, adst1, N);
  gat_edge_max1<<<nb(E * 8, B), B, 0, stream>>>(ei, asrc1, adst1, emax1, E);
  gat_edge_exp1<<<nb(E * 8, B), B, 0, stream>>>(ei, asrc1, adst1, emax1, ex1, den1, E);
  gat_edge_scatter1<<<nb(E * 8, B), B, 0, stream>>>(ei, ex1, den1, h1, out1, E);
  gat_elu_bias<<<nb(N * 64, B), B, 0, stream>>>(out1, bias1, N);

  // Layer 2
  gat_gemm2_wmma<<<dim3((N + 63) / 64), dim3(32, 4), 0, stream>>>(out1, W2, h2, N);
  gat_att2<<<nb(N, B), B, 0, stream>>>(h2, att_src2, att_dst2, asrc2, adst2, N);
  gat_edge_max2<<<nb(E, B), B, 0, stream>>>(ei, asrc2, adst2, emax2, E);
  gat_edge_exp2<<<nb(E, B), B, 0, stream>>>(ei, asrc2, adst2, emax2, ex2, den2, E);
  gat_edge_scatter2<<<nb(E * 5, B), B, 0, stream>>>(ei, ex2, den2, h2, out, E);
  gat_bias2<<<nb(N * 40, B), B, 0, stream>>>(out, bias2, N);
}